// IntraCBLoss_50139448213777
// MI455X (gfx1250) — compile-verified
//
#include <hip/hip_runtime.h>
#include <hip/hip_bf16.h>

// IntraCB loss, MI455X (gfx1250, wave32).
// Bandwidth-bound streaming reduction: ~201 MB read once -> ~8.6 us floor @ 23.3 TB/s.
// Sort in the reference is replaced by an exact count: kth-largest <= 0.1  <=>  count(>0.1) < k.
// Streams don't fit the 192 MB L2 and are touched once -> non-temporal (TH_NT) loads.

#define ICB_B      16
#define ICB_HW     (1024 * 1024)
#define ICB_NVEC   (ICB_HW / 4)          // 262144 float4 per stream per batch
#define ICB_GX     256
#define ICB_BLOCK  256
#define ICB_STRIDE (ICB_GX * ICB_BLOCK)  // 65536 threads per batch
#define ICB_ITERS  (ICB_NVEC / ICB_STRIDE) // exactly 4, no remainder
#define ICB_NACC   10   // s_df n_df s_sf n_sf s_db n_db s_sb n_sb c_f c_b

typedef __attribute__((ext_vector_type(4))) float v4f;

__device__ __forceinline__ void icb_pixel(float x, float g, float (&a)[ICB_NACC]) {
    // q = clamp(sigmoid(fg ?  (p1-p0) : -(p1-p0)))   (== pc for fg, 1-pc for bg)
    // loss = -log(q);  "difficult" bin: q <= T
    float u   = __builtin_fmaf(-2.0f * g, x, x);     // (1-2g)*x  -> exp argument
    float e   = __expf(u);
    float q   = __builtin_amdgcn_rcpf(1.0f + e);     // v_rcp_f32
    q         = fminf(fmaxf(q, 1e-7f), 1.0f - 1e-7f);
    float loss = -__logf(q);

    bool  dm  = (q <= 0.9f);
    float ld  = dm ? loss : 0.0f;
    float lnd = loss - ld;
    float cd  = dm ? 1.0f : 0.0f;
    float cnd = 1.0f - cd;
    float fv  = 1.0f - q;                            // fg: gt*(1-pc);  bg: (1-gt)*pc
    float cfar = (fv > 0.1f) ? 1.0f : 0.0f;          // counts toward order-stat
    float gb  = 1.0f - g;                            // g is exactly 0.0 or 1.0

    a[0] = __builtin_fmaf(ld,   g,  a[0]);   // s_df
    a[1] = __builtin_fmaf(cd,   g,  a[1]);   // n_df
    a[2] = __builtin_fmaf(lnd,  g,  a[2]);   // s_sf
    a[3] = __builtin_fmaf(cnd,  g,  a[3]);   // n_sf
    a[4] = __builtin_fmaf(ld,   gb, a[4]);   // s_db
    a[5] = __builtin_fmaf(cd,   gb, a[5]);   // n_db
    a[6] = __builtin_fmaf(lnd,  gb, a[6]);   // s_sb
    a[7] = __builtin_fmaf(cnd,  gb, a[7]);   // n_sb
    a[8] = __builtin_fmaf(cfar, g,  a[8]);   // c_f : fg count of (1-pc) > 0.1
    a[9] = __builtin_fmaf(cfar, gb, a[9]);   // c_b : bg count of pc > 0.1
}

__global__ void icb_init(float* __restrict__ ws) {
    int i = threadIdx.x;
    if (i < ICB_B * ICB_NACC) ws[i] = 0.0f;
}

__global__ __launch_bounds__(ICB_BLOCK)
void icb_reduce(const float* __restrict__ probs, const float* __restrict__ gt,
                float* __restrict__ ws) {
    const int b = blockIdx.y;
    const v4f* __restrict__ p0 =
        reinterpret_cast<const v4f*>(probs + (size_t)b * (2u * ICB_HW));
    const v4f* __restrict__ p1 =
        reinterpret_cast<const v4f*>(probs + (size_t)b * (2u * ICB_HW) + ICB_HW);
    const v4f* __restrict__ gv =
        reinterpret_cast<const v4f*>(gt + (size_t)b * ICB_HW);

    float a[ICB_NACC];
#pragma unroll
    for (int j = 0; j < ICB_NACC; ++j) a[j] = 0.0f;

    const int tid = blockIdx.x * ICB_BLOCK + threadIdx.x;   // < ICB_STRIDE

    // Static trip count: all 3*ICB_ITERS global_load_b128 (TH_NT) issue up front.
    v4f A[ICB_ITERS], Bv[ICB_ITERS], G[ICB_ITERS];
#pragma unroll
    for (int k = 0; k < ICB_ITERS; ++k) {
        const int i = tid + k * ICB_STRIDE;
        A[k]  = __builtin_nontemporal_load(&p0[i]);
        Bv[k] = __builtin_nontemporal_load(&p1[i]);
        G[k]  = __builtin_nontemporal_load(&gv[i]);
    }
#pragma unroll
    for (int k = 0; k < ICB_ITERS; ++k) {
        icb_pixel(Bv[k].x - A[k].x, G[k].x, a);   // x = p1 - p0 (2-class softmax)
        icb_pixel(Bv[k].y - A[k].y, G[k].y, a);
        icb_pixel(Bv[k].z - A[k].z, G[k].z, a);
        icb_pixel(Bv[k].w - A[k].w, G[k].w, a);
    }

    // wave32 cross-lane reduction, then LDS combine, then 10 global atomics/block
    __shared__ float sacc[ICB_NACC];
    if (threadIdx.x < ICB_NACC) sacc[threadIdx.x] = 0.0f;
    __syncthreads();

    const int lane = threadIdx.x & 31;
#pragma unroll
    for (int j = 0; j < ICB_NACC; ++j) {
        float v = a[j];
        v += __shfl_down(v, 16, 32);
        v += __shfl_down(v,  8, 32);
        v += __shfl_down(v,  4, 32);
        v += __shfl_down(v,  2, 32);
        v += __shfl_down(v,  1, 32);
        if (lane == 0) atomicAdd(&sacc[j], v);   // ds_add_f32
    }
    __syncthreads();
    if (threadIdx.x < ICB_NACC)
        atomicAdd(&ws[b * ICB_NACC + threadIdx.x], sacc[threadIdx.x]);
}

__global__ void icb_final(const float* __restrict__ ws,
                          const int* __restrict__ gt_f_num,
                          const int* __restrict__ bs,
                          float* __restrict__ out, int out_size) {
    if (threadIdx.x != 0 || blockIdx.x != 0) return;
    float total = 0.0f;
    int sum_f = 0, sum_b = 0;
    for (int b = 0; b < ICB_B; ++b) {
        const float* w = ws + b * ICB_NACC;
        total += w[0] / (w[1] + 1e-16f)    // term(m_df)
               + w[2] / (w[3] + 1e-16f)    // term(m_sf)
               + w[4] / (w[5] + 1e-16f)    // term(m_db)
               + w[6] / (w[7] + 1e-16f);   // term(m_sb)
        float k = (float)gt_f_num[b];
        // kth-largest <= 0.1  <=>  count(>0.1) < k
        if (w[8] < k) sum_f++;
        if (w[9] < k) sum_b++;
    }
    total *= (1.0f / (float)ICB_B);
    int trans = (sum_b >= ICB_B / 2) && (sum_f >= ICB_B / 2) && (bs[0] == ICB_B);
    out[0] = total;
    if (out_size > 1) out[1] = trans ? 1.0f : 0.0f;
}

extern "C" void kernel_launch(void* const* d_in, const int* in_sizes, int n_in,
                              void* d_out, int out_size, void* d_ws, size_t ws_size,
                              hipStream_t stream) {
    (void)in_sizes; (void)n_in; (void)ws_size;
    const float* probs    = (const float*)d_in[0];   // [16,2,1024,1024] f32
    const float* gt       = (const float*)d_in[1];   // [16,1024,1024]  f32
    const int*   gt_f_num = (const int*)d_in[2];     // [16] i32
    const int*   bs       = (const int*)d_in[3];     // [1]  i32
    float*       out      = (float*)d_out;
    float*       ws       = (float*)d_ws;            // 160 floats used

    icb_init<<<1, ICB_B * ICB_NACC, 0, stream>>>(ws);
    dim3 grid(ICB_GX, ICB_B);
    icb_reduce<<<grid, ICB_BLOCK, 0, stream>>>(probs, gt, ws);
    icb_final<<<1, 32, 0, stream>>>(ws, gt_f_num, bs, out, out_size);
}